// MeshNorms_21852793602422
// MI455X (gfx1250) — compile-verified
//
#include <hip/hip_runtime.h>
#include <math.h>

// Mesh vertex normals, MI455X (gfx1250, wave32).
// Bandwidth-bound gather workload: ~0.2 GFLOP vs ~150-200 MB traffic
// => ~8 us at 23.3 TB/s HBM. No GEMM structure => WMMA not applicable.
// Strategy:
//  - two passes; 50 MB face-normal intermediate kept L2-resident (192 MB L2)
//  - streaming arrays (faces, normmap, output) use non-temporal cache policy
//    so they don't evict the reused intermediate/verts from L2
//  - both batches per thread => index arrays read once, not BATCH times
//  - global_prefetch_b8 (gfx1250) pulls the index streams into L2 ahead
//  - DEG=6 specialization vectorizes the normmap row loads (8B-aligned rows)

namespace {

constexpr int kBatch = 2;

__device__ __forceinline__ void cross_edges(
    float ax, float ay, float az,
    float bx, float by, float bz,
    float cx_, float cy_, float cz_,
    float& ox, float& oy, float& oz) {
  // v1 = a - b ; v2 = a - c ; out = cross(v1, v2)
  float e1x = ax - bx, e1y = ay - by, e1z = az - bz;
  float e2x = ax - cx_, e2y = ay - cy_, e2z = az - cz_;
  ox = e1y * e2z - e1z * e2y;
  oy = e1z * e2x - e1x * e2z;
  oz = e1x * e2y - e1y * e2x;
}

// Pass 1: one thread per face (plus one for the zero-pad row).
// fn layout: [BATCH][F+1][3] floats. fn is written RT (reused in pass 2).
__global__ void face_normals_kernel(const float* __restrict__ verts,
                                    const int* __restrict__ faces,
                                    float* __restrict__ fn,
                                    int F, int V) {
  int f = blockIdx.x * blockDim.x + threadIdx.x;
  if (f > F) return;

  size_t batch_stride = (size_t)(F + 1) * 3;

  if (f == F) {
    // Zero-pad row (index F) for both batches; rewritten every launch
    // (workspace is poisoned by the harness).
#pragma unroll
    for (int b = 0; b < kBatch; ++b) {
      float* o = fn + (size_t)b * batch_stride + (size_t)F * 3;
      o[0] = 0.0f; o[1] = 0.0f; o[2] = 0.0f;
    }
    return;
  }

  // gfx1250 L2 prefetch of the streaming index array (bounds-guarded:
  // WGP-scope prefetches are non-speculative).
  if (f + 16384 < F) {
    __builtin_prefetch(faces + 3 * ((size_t)f + 16384), 0, 1);
  }

  // faces is read exactly once -> non-temporal so it doesn't pollute L2.
  int i0 = __builtin_nontemporal_load(faces + 3 * (size_t)f + 0);
  int i1 = __builtin_nontemporal_load(faces + 3 * (size_t)f + 1);
  int i2 = __builtin_nontemporal_load(faces + 3 * (size_t)f + 2);

#pragma unroll
  for (int b = 0; b < kBatch; ++b) {
    const float* vb = verts + (size_t)b * V * 3;  // verts: RT (heavy reuse)
    const float* p0 = vb + 3 * (size_t)i0;
    const float* p1 = vb + 3 * (size_t)i1;
    const float* p2 = vb + 3 * (size_t)i2;
    float cx, cy, cz;
    cross_edges(p0[0], p0[1], p0[2],
                p1[0], p1[1], p1[2],
                p2[0], p2[1], p2[2], cx, cy, cz);
    float n = sqrtf(cx * cx + cy * cy + cz * cz);
    float* o = fn + (size_t)b * batch_stride + 3 * (size_t)f;
    o[0] = cx / n;   // RT store: consumed by pass 2 out of L2
    o[1] = cy / n;
    o[2] = cz / n;
  }
}

// Pass 2: one thread per vertex; gathers deg face normals (L2-resident
// intermediate) for both batches, sums in normmap-column order, normalizes.
// DEG > 0: compile-time degree (unrolled, vectorizable NT index loads).
// DEG == 0: runtime degree fallback.
template <int DEG>
__global__ void vert_normals_kernel(const int* __restrict__ normmap,
                                    const float* __restrict__ fn,
                                    float* __restrict__ out,
                                    int V, int F, int deg_rt) {
  const int deg = (DEG > 0) ? DEG : deg_rt;
  int v = blockIdx.x * blockDim.x + threadIdx.x;
  if (v >= V) return;

  const int* row = normmap + (size_t)v * deg;
  if (v + 16384 < V) {
    __builtin_prefetch(normmap + ((size_t)v + 16384) * deg, 0, 1);
  }

  size_t batch_stride = (size_t)(F + 1) * 3;
  float s0x = 0.0f, s0y = 0.0f, s0z = 0.0f;
  float s1x = 0.0f, s1y = 0.0f, s1z = 0.0f;

  if constexpr (DEG > 0) {
    // Read-once index row: non-temporal, unrolled so the backend can merge
    // the contiguous b32 loads (row base is 8-byte aligned for DEG=6).
    int idx[DEG];
#pragma unroll
    for (int k = 0; k < DEG; ++k) idx[k] = __builtin_nontemporal_load(row + k);
#pragma unroll
    for (int k = 0; k < DEG; ++k) {
      size_t off = 3 * (size_t)idx[k];  // idx == F hits the zero-pad row
      const float* p0 = fn + off;
      const float* p1 = fn + batch_stride + off;
      s0x += p0[0]; s0y += p0[1]; s0z += p0[2];
      s1x += p1[0]; s1y += p1[1]; s1z += p1[2];
    }
  } else {
    for (int k = 0; k < deg; ++k) {
      size_t off = 3 * (size_t)__builtin_nontemporal_load(row + k);
      const float* p0 = fn + off;
      const float* p1 = fn + batch_stride + off;
      s0x += p0[0]; s0y += p0[1]; s0z += p0[2];
      s1x += p1[0]; s1y += p1[1]; s1z += p1[2];
    }
  }

  float n0 = sqrtf(s0x * s0x + s0y * s0y + s0z * s0z);
  float n1 = sqrtf(s1x * s1x + s1y * s1y + s1z * s1z);

  // Output is write-once / never re-read: non-temporal stores keep it from
  // evicting the fn intermediate while pass 2 is still running.
  float* o0 = out + 3 * (size_t)v;
  float* o1 = out + (size_t)V * 3 + 3 * (size_t)v;
  __builtin_nontemporal_store(s0x / n0, o0 + 0);
  __builtin_nontemporal_store(s0y / n0, o0 + 1);
  __builtin_nontemporal_store(s0z / n0, o0 + 2);
  __builtin_nontemporal_store(s1x / n1, o1 + 0);
  __builtin_nontemporal_store(s1y / n1, o1 + 1);
  __builtin_nontemporal_store(s1z / n1, o1 + 2);
}

// Fallback: fused single pass (no workspace) — recomputes face normals per
// vertex; verts (12.6 MB/batch) stay L2-resident so redundant gathers are
// cheap. Used only if ws_size can't hold the intermediate.
__global__ void fused_kernel(const float* __restrict__ verts,
                             const int* __restrict__ faces,
                             const int* __restrict__ normmap,
                             float* __restrict__ out,
                             int V, int F, int deg) {
  int v = blockIdx.x * blockDim.x + threadIdx.x;
  if (v >= V) return;

  const int* row = normmap + (size_t)v * deg;
  float acc[kBatch][3] = {};

  for (int k = 0; k < deg; ++k) {
    int f = __builtin_nontemporal_load(row + k);
    if (f >= F) continue;  // pad entry contributes zero
    int i0 = faces[3 * (size_t)f + 0];
    int i1 = faces[3 * (size_t)f + 1];
    int i2 = faces[3 * (size_t)f + 2];
#pragma unroll
    for (int b = 0; b < kBatch; ++b) {
      const float* vb = verts + (size_t)b * V * 3;
      const float* p0 = vb + 3 * (size_t)i0;
      const float* p1 = vb + 3 * (size_t)i1;
      const float* p2 = vb + 3 * (size_t)i2;
      float cx, cy, cz;
      cross_edges(p0[0], p0[1], p0[2],
                  p1[0], p1[1], p1[2],
                  p2[0], p2[1], p2[2], cx, cy, cz);
      float n = sqrtf(cx * cx + cy * cy + cz * cz);
      acc[b][0] += cx / n;
      acc[b][1] += cy / n;
      acc[b][2] += cz / n;
    }
  }

#pragma unroll
  for (int b = 0; b < kBatch; ++b) {
    float n = sqrtf(acc[b][0] * acc[b][0] + acc[b][1] * acc[b][1] +
                    acc[b][2] * acc[b][2]);
    float* o = out + (size_t)b * V * 3 + 3 * (size_t)v;
    __builtin_nontemporal_store(acc[b][0] / n, o + 0);
    __builtin_nontemporal_store(acc[b][1] / n, o + 1);
    __builtin_nontemporal_store(acc[b][2] / n, o + 2);
  }
}

}  // namespace

extern "C" void kernel_launch(void* const* d_in, const int* in_sizes, int n_in,
                              void* d_out, int out_size, void* d_ws, size_t ws_size,
                              hipStream_t stream) {
  const float* verts   = (const float*)d_in[0];  // (B, V, 3) f32
  const int*   faces   = (const int*)d_in[1];    // (F, 3)    i32
  const int*   normmap = (const int*)d_in[2];    // (V, deg)  i32
  float*       out     = (float*)d_out;          // (B, V, 3) f32

  const int F   = in_sizes[1] / 3;
  const int V   = out_size / (kBatch * 3);
  const int deg = in_sizes[2] / V;

  const int threads = 256;  // 8 wave32 waves per block
  const size_t ws_need = (size_t)kBatch * (size_t)(F + 1) * 3 * sizeof(float);

  if (ws_size >= ws_need) {
    float* fn = (float*)d_ws;
    int nf = F + 1;  // includes zero-pad row
    face_normals_kernel<<<(nf + threads - 1) / threads, threads, 0, stream>>>(
        verts, faces, fn, F, V);
    int vblocks = (V + threads - 1) / threads;
    if (deg == 6) {
      // Grid mesh: interior vertices touch exactly 6 triangles.
      vert_normals_kernel<6><<<vblocks, threads, 0, stream>>>(
          normmap, fn, out, V, F, deg);
    } else if (deg == 4) {
      vert_normals_kernel<4><<<vblocks, threads, 0, stream>>>(
          normmap, fn, out, V, F, deg);
    } else if (deg == 8) {
      vert_normals_kernel<8><<<vblocks, threads, 0, stream>>>(
          normmap, fn, out, V, F, deg);
    } else {
      vert_normals_kernel<0><<<vblocks, threads, 0, stream>>>(
          normmap, fn, out, V, F, deg);
    }
  } else {
    fused_kernel<<<(V + threads - 1) / threads, threads, 0, stream>>>(
        verts, faces, normmap, out, V, F, deg);
  }
}